// FeatExtractor_14551349199514
// MI455X (gfx1250) — compile-verified
//
#include <hip/hip_runtime.h>
#include <hip/hip_bf16.h>
#include <stdint.h>

// ---------------------------------------------------------------------------
// ROI feature extractor (BN-fused crop_and_resize, 3 FPN levels, concat).
// Gather/memory-bound: use CDNA5 async global->LDS loads (ASYNCcnt) to keep
// scattered taps in flight with a 2-stage software pipeline.
// ---------------------------------------------------------------------------

#define N_CH_TOTAL 1792      // 256 + 512 + 1024
#define CROP 7
#define SPAN (256 * 7)       // outputs per block (256 threads x 7 pipelined iters)

typedef __attribute__((address_space(1))) int  g_int;
typedef __attribute__((address_space(3))) int  l_int;

// ---- CDNA5 async global->LDS b32 gather -----------------------------------
__device__ __forceinline__ void async_gather_b32(const float* g, float* lds) {
#if defined(__has_builtin) && __has_builtin(__builtin_amdgcn_global_load_async_to_lds_b32)
    __builtin_amdgcn_global_load_async_to_lds_b32(
        (g_int*)(uintptr_t)g,
        (l_int*)(uintptr_t)(uint32_t)(uintptr_t)lds,
        0, 0);
#else
    uint32_t loff = (uint32_t)(uintptr_t)lds;          // low 32 bits = LDS byte addr
    uint64_t gaddr = (uint64_t)(uintptr_t)g;
    asm volatile("global_load_async_to_lds_b32 %0, %1, off"
                 :: "v"(loff), "v"(gaddr)
                 : "memory");
#endif
}

template <int N>
__device__ __forceinline__ void wait_asynccnt() {
#if defined(__has_builtin) && __has_builtin(__builtin_amdgcn_s_wait_asynccnt)
    __builtin_amdgcn_s_wait_asynccnt(N);
#else
    asm volatile("s_wait_asynccnt %0" :: "n"(N) : "memory");
#endif
    asm volatile("" ::: "memory");   // keep LDS reads after the wait
}

// ---- BN affine precompute: inv[c] = g/sqrt(v+eps), off[c] = b - m*inv -----
__global__ void bn_precompute_kernel(
    const float* __restrict__ g0, const float* __restrict__ b0,
    const float* __restrict__ m0, const float* __restrict__ v0,
    const float* __restrict__ g1, const float* __restrict__ b1,
    const float* __restrict__ m1, const float* __restrict__ v1,
    const float* __restrict__ g2, const float* __restrict__ b2,
    const float* __restrict__ m2, const float* __restrict__ v2,
    float* __restrict__ inv, float* __restrict__ off) {
    int c = blockIdx.x * blockDim.x + threadIdx.x;
    if (c >= N_CH_TOTAL) return;
    const float *g, *b, *m, *v; int ci;
    if (c < 256)      { g = g0; b = b0; m = m0; v = v0; ci = c; }
    else if (c < 768) { g = g1; b = b1; m = m1; v = v1; ci = c - 256; }
    else              { g = g2; b = b2; m = m2; v = v2; ci = c - 768; }
    float iv = g[ci] * rsqrtf(v[ci] + 1e-5f);
    inv[c] = iv;
    off[c] = b[ci] - m[ci] * iv;
}

// ---- Main per-level kernel -------------------------------------------------
// grid = (C*49/SPAN, nRois); block = 256. Each thread produces 7 outputs,
// double-buffering its 4 bilinear taps through LDS via async gathers.
__global__ void __launch_bounds__(256)
roi_level_kernel(const float* __restrict__ feat,
                 const float* __restrict__ boxes,
                 const int*   __restrict__ inds,
                 const float* __restrict__ invArr,
                 const float* __restrict__ offArr,
                 float* __restrict__ out,
                 int C, int H, int W,
                 float invStride, int cOff) {
    const int roi  = blockIdx.y;
    const int tid  = threadIdx.x;
    const int base = blockIdx.x * SPAN;         // flat offset within this level's section

    // Uniform per-ROI values (scalarized by the compiler: roi is blockIdx.y).
    const float bx1 = boxes[roi * 4 + 0] * invStride;
    const float by1 = boxes[roi * 4 + 1] * invStride;
    const float bx2 = boxes[roi * 4 + 2] * invStride;
    const float by2 = boxes[roi * 4 + 3] * invStride;
    const int   bimg = inds[roi];
    const float dx = (bx2 - bx1) * (1.0f / 6.0f);
    const float dy = (by2 - by1) * (1.0f / 6.0f);
    const float Hm1 = (float)(H - 1);
    const float Wm1 = (float)(W - 1);
    const float* fb = feat + (size_t)bimg * (size_t)C * H * W;

    __shared__ __align__(16) float stage[2][256 * 4];

    float  wxA[7], wyA[7], invA[7], offA[7];
    bool   validA[7];
    size_t outIdxA[7];

    auto issue = [&](int it, int buf) {
        const int flat = base + it * 256 + tid;
        const int c = flat / 49;
        const int p = flat - c * 49;
        const int i = p / 7;
        const int j = p - i * 7;
        const float Y = by1 + dy * (float)i;
        const float X = bx1 + dx * (float)j;
        const float y0f = floorf(Y), x0f = floorf(X);
        int y0 = (int)y0f; y0 = y0 < 0 ? 0 : (y0 > H - 1 ? H - 1 : y0);
        int x0 = (int)x0f; x0 = x0 < 0 ? 0 : (x0 > W - 1 ? W - 1 : x0);
        const int y1i = (y0 + 1 > H - 1) ? H - 1 : y0 + 1;
        const int x1i = (x0 + 1 > W - 1) ? W - 1 : x0 + 1;
        const float* plane = fb + (size_t)c * H * W;
        const float* r0 = plane + (size_t)y0  * W;
        const float* r1 = plane + (size_t)y1i * W;
        float* l = &stage[buf][tid * 4];
        async_gather_b32(r0 + x0,  l + 0);
        async_gather_b32(r0 + x1i, l + 1);
        async_gather_b32(r1 + x0,  l + 2);
        async_gather_b32(r1 + x1i, l + 3);
        wxA[it] = X - x0f;
        wyA[it] = Y - y0f;
        validA[it] = (Y >= 0.0f) & (Y <= Hm1) & (X >= 0.0f) & (X <= Wm1);
        const int cg = cOff + c;
        invA[it] = invArr[cg];
        offA[it] = offArr[cg];
        outIdxA[it] = (size_t)roi * (size_t)(N_CH_TOTAL * 49) + (size_t)cg * 49 + p;
    };

    issue(0, 0);
#pragma unroll
    for (int it = 0; it < 7; ++it) {
        if (it < 6) {
            issue(it + 1, (it + 1) & 1);
            wait_asynccnt<4>();     // this wave: 8 outstanding -> first 4 (stage it) done
        } else {
            wait_asynccnt<0>();     // drain the last stage
        }
        const float4 v = *(const float4*)&stage[it & 1][tid * 4];
        const float wy = wyA[it], wx = wxA[it];
        const float omy = 1.0f - wy, omx = 1.0f - wx;
        const float interp = v.x * omy * omx + v.y * omy * wx +
                             v.z * wy  * omx + v.w * wy  * wx;
        const float r = validA[it] ? fmaf(interp, invA[it], offA[it]) : 0.0f;
        out[outIdxA[it]] = r;
    }
}

extern "C" void kernel_launch(void* const* d_in, const int* in_sizes, int n_in,
                              void* d_out, int out_size, void* d_ws, size_t ws_size,
                              hipStream_t stream) {
    // setup_inputs() dict order:
    // 0 feat0, 1 feat1, 2 feat2, 3 roi_boxes, 4 roi_inds,
    // 5..8 bn0 g/b/m/v, 9..12 bn1 g/b/m/v, 13..16 bn2 g/b/m/v
    const float* feat0 = (const float*)d_in[0];
    const float* feat1 = (const float*)d_in[1];
    const float* feat2 = (const float*)d_in[2];
    const float* boxes = (const float*)d_in[3];
    const int*   inds  = (const int*)d_in[4];

    float* inv = (float*)d_ws;
    float* off = inv + N_CH_TOTAL;
    float* out = (float*)d_out;
    const int nRois = in_sizes[4];

    bn_precompute_kernel<<<(N_CH_TOTAL + 255) / 256, 256, 0, stream>>>(
        (const float*)d_in[5],  (const float*)d_in[6],
        (const float*)d_in[7],  (const float*)d_in[8],
        (const float*)d_in[9],  (const float*)d_in[10],
        (const float*)d_in[11], (const float*)d_in[12],
        (const float*)d_in[13], (const float*)d_in[14],
        (const float*)d_in[15], (const float*)d_in[16],
        inv, off);

    // blocks_x = C*49 / SPAN  (exact: 7 / 14 / 28)
    roi_level_kernel<<<dim3(7,  nRois), 256, 0, stream>>>(
        feat0, boxes, inds, inv, off, out, 256, 200, 200, 1.0f / 4.0f, 0);
    roi_level_kernel<<<dim3(14, nRois), 256, 0, stream>>>(
        feat1, boxes, inds, inv, off, out, 512, 100, 100, 1.0f / 8.0f, 256);
    roi_level_kernel<<<dim3(28, nRois), 256, 0, stream>>>(
        feat2, boxes, inds, inv, off, out, 1024, 50, 50, 1.0f / 16.0f, 768);
}